// SegDropout_73864847556934
// MI455X (gfx1250) — compile-verified
//
#include <hip/hip_runtime.h>
#include <math.h>

#define BATCH   8192
#define FEATS   4096
#define NCHUNK  16
#define ROWS_PER_CHUNK (BATCH / NCHUNK)   // 512
#define RATE    0.1f
#define ANNEAL  0.6f                      // 0.5 + 0.1

typedef __attribute__((ext_vector_type(2))) float v2f;
typedef __attribute__((ext_vector_type(4))) float v4f;
typedef __attribute__((ext_vector_type(8))) float v8f;

// ---------------------------------------------------------------------------
// Kernel 1: column sums via V_WMMA_F32_16X16X4_F32 with an all-ones A matrix.
//   C[m][n] += sum_k A[m][k]*B[k][n]; A == ones  =>  every row of C is the
//   column sum of B. Each wave owns a 16-feature tile and a 512-row chunk.
//   B layout (32-bit B, 4x16): VGPR0 lanes0-15 = K0, lanes16-31 = K2;
//                              VGPR1 lanes0-15 = K1, lanes16-31 = K3.
//   (With A==ones the result is layout-invariant anyway.)
// ---------------------------------------------------------------------------
__global__ __launch_bounds__(32)
void segdrop_colsum_wmma(const float* __restrict__ in, float* __restrict__ part) {
    const int lane  = threadIdx.x;                 // 0..31, one wave per block
    const int fbase = blockIdx.x * 16;             // 256 feature tiles
    const int chunk = blockIdx.y;                  // 16 batch chunks
    const int f     = fbase + (lane & 15);
    const int rsel  = (lane >> 4) * 2;             // rows K0/K1 vs K2/K3

    v2f a; a.x = 1.0f; a.y = 1.0f;                 // all-ones 16x4 A matrix
    v8f c = {};

    const float* p = in + (size_t)(chunk * ROWS_PER_CHUNK + rsel) * FEATS + f;
    #pragma unroll 4
    for (int it = 0; it < ROWS_PER_CHUNK / 4; ++it) {
        v2f b;
        b.x = p[0];            // row r+rsel   (temporal: seed 192MB L2 residency)
        b.y = p[FEATS];        // row r+rsel+1
        c = __builtin_amdgcn_wmma_f32_16x16x4_f32(
                /*neg_a=*/false, a, /*neg_b=*/false, b,
                /*c_mod=*/(short)0, c, /*reuse_a=*/false, /*reuse_b=*/false);
        p += 4 * (size_t)FEATS;
    }

    // C row M=0 lives in VGPR0 of lanes 0..15 (N = lane). Deterministic store.
    if (lane < 16) part[chunk * FEATS + f] = c[0];
}

// ---------------------------------------------------------------------------
// Kernel 2: one block. Reduce partials -> mean, min/max, binning, LDS
// histogram, annealed dropout rate, final per-feature scale = 1 + sigma*noise.
// ---------------------------------------------------------------------------
__global__ __launch_bounds__(1024)
void segdrop_stats(const float* __restrict__ part,
                   const float* __restrict__ noise,
                   float* __restrict__ scale) {
    __shared__ float xagg[FEATS];
    __shared__ int   hist[FEATS + 1];
    __shared__ float smin[1024];
    __shared__ float smax[1024];

    const int tid = threadIdx.x;
    float lmin =  3.402823466e38f;
    float lmax = -3.402823466e38f;

    for (int f = tid; f < FEATS; f += 1024) {
        float s = 0.0f;
        #pragma unroll
        for (int c = 0; c < NCHUNK; ++c) s += part[c * FEATS + f];
        float m = s * (1.0f / (float)BATCH);
        xagg[f] = m;
        lmin = fminf(lmin, m);
        lmax = fmaxf(lmax, m);
    }
    smin[tid] = lmin; smax[tid] = lmax;
    __syncthreads();
    for (int s = 512; s > 0; s >>= 1) {
        if (tid < s) {
            smin[tid] = fminf(smin[tid], smin[tid + s]);
            smax[tid] = fmaxf(smax[tid], smax[tid + s]);
        }
        __syncthreads();
    }
    const float xmin = smin[0], xmax = smax[0];
    const float inv  = (float)FEATS / (xmax - xmin);

    for (int i = tid; i <= FEATS; i += 1024) hist[i] = 0;
    __syncthreads();

    for (int f = tid; f < FEATS; f += 1024) {
        int bin = (int)((xagg[f] - xmin) * inv);   // [0, FEATS]
        atomicAdd(&hist[bin], 1);                  // integer: deterministic
    }
    __syncthreads();

    const float logR = logf(RATE);
    for (int f = tid; f < FEATS; f += 1024) {
        int   bin = (int)((xagg[f] - xmin) * inv);
        float cnt = (float)hist[bin];
        float dr  = (cnt == 1.0f) ? RATE : expf(logR / (ANNEAL * cnt));
        float sg  = sqrtf(dr / (1.0f - dr));
        scale[f]  = 1.0f + sg * noise[f];
    }
}

// ---------------------------------------------------------------------------
// Kernel 3: out = in * scale[f]. Input re-read is L2-resident (192MB L2 holds
// the whole 128MB tensor seeded by kernel 1); output streamed with
// non-temporal stores so it does not evict the resident input.
// ---------------------------------------------------------------------------
__global__ __launch_bounds__(256)
void segdrop_apply(const v4f* __restrict__ in,
                   const float* __restrict__ scale,
                   v4f* __restrict__ out, int n4) {
    int idx = blockIdx.x * 256 + threadIdx.x;
    if (idx >= n4) return;
    v4f s = ((const v4f*)scale)[idx & (FEATS / 4 - 1)];
    v4f v = in[idx];                       // temporal load -> L2 hit
    v4f r;
    r.x = v.x * s.x; r.y = v.y * s.y; r.z = v.z * s.z; r.w = v.w * s.w;
    __builtin_nontemporal_store(r, &out[idx]);   // TH=NT store, spare the L2
}

extern "C" void kernel_launch(void* const* d_in, const int* in_sizes, int n_in,
                              void* d_out, int out_size, void* d_ws, size_t ws_size,
                              hipStream_t stream) {
    const float* in    = (const float*)d_in[0];   // [8192*4096] f32
    const float* noise = (const float*)d_in[1];   // [4096] f32
    float*       out   = (float*)d_out;           // [8192*4096] f32

    float* part  = (float*)d_ws;                  // [16][4096] partial col sums
    float* scale = part + (size_t)NCHUNK * FEATS; // [4096] final scale

    dim3 g1(FEATS / 16, NCHUNK);                  // 256 tiles x 16 chunks
    segdrop_colsum_wmma<<<g1, 32, 0, stream>>>(in, part);

    segdrop_stats<<<1, 1024, 0, stream>>>(part, noise, scale);

    int n4 = BATCH * FEATS / 4;                   // 8,388,608 float4s
    segdrop_apply<<<n4 / 256, 256, 0, stream>>>((const v4f*)in, scale,
                                                (v4f*)out, n4);
}